// ContextLayer_7052336300197
// MI455X (gfx1250) — compile-verified
//
#include <hip/hip_runtime.h>
#include <hip/hip_bf16.h>

#define NPTS 32768
#define NGRP 1024

typedef __attribute__((ext_vector_type(2))) float v2f;
typedef __attribute__((ext_vector_type(8))) float v8f;

// ---------------------------------------------------------------------------
// Kernel 1: Farthest point sampling (exact int arithmetic, first-index ties)
// One workgroup of 1024 threads per batch. Each thread owns 32 points
// (strided p = j*1024 + t) in registers: packed coords + running min dist.
// ---------------------------------------------------------------------------
__global__ __launch_bounds__(1024) void fps_kernel(
    const int* __restrict__ coords, const float* __restrict__ features,
    int* __restrict__ gidx, unsigned* __restrict__ gcoord,
    float* __restrict__ gfeat) {
  const int b = blockIdx.x;
  const int t = threadIdx.x;
  const int* cb = coords + (long)b * NPTS * 3;

  unsigned pc[32];
  int dmin[32];
#pragma unroll
  for (int j = 0; j < 32; ++j) {
    int p = j * 1024 + t;
    unsigned x = (unsigned)cb[p * 3 + 0];
    unsigned y = (unsigned)cb[p * 3 + 1];
    unsigned z = (unsigned)cb[p * 3 + 2];
    pc[j] = x | (y << 10) | (z << 20);
    dmin[j] = 0x7fffffff;
  }

  __shared__ unsigned long long skey[32];
  __shared__ int swin;
  __shared__ unsigned scent;
  __shared__ unsigned short sgidx[NGRP];

  if (t == 0) scent = pc[0];  // thread 0 owns point 0
  int farthest = 0;
  __syncthreads();

  for (int i = 0; i < NGRP; ++i) {
    unsigned cent = scent;
    if (t == 0) {
      sgidx[i] = (unsigned short)farthest;
      gidx[b * NGRP + i] = farthest;
      gcoord[b * NGRP + i] = cent;
    }
    int cx = (int)(cent & 1023u);
    int cy = (int)((cent >> 10) & 1023u);
    int cz = (int)(cent >> 20);

    unsigned long long best = 0ull;
#pragma unroll
    for (int j = 0; j < 32; ++j) {
      int dx = (int)(pc[j] & 1023u) - cx;
      int dy = (int)((pc[j] >> 10) & 1023u) - cy;
      int dz = (int)(pc[j] >> 20) - cz;
      int d = dx * dx + dy * dy + dz * dz;
      int dm = dmin[j];
      dm = d < dm ? d : dm;
      dmin[j] = dm;
      unsigned idx = (unsigned)(j * 1024 + t);
      unsigned long long key =
          ((unsigned long long)(unsigned)dm << 32) | (unsigned)(~idx);
      best = key > best ? key : best;
    }
    // wave32 reduce (max key)
#pragma unroll
    for (int s = 16; s > 0; s >>= 1) {
      unsigned long long o = __shfl_xor(best, s, 32);
      best = o > best ? o : best;
    }
    if ((t & 31) == 0) skey[t >> 5] = best;
    __syncthreads();
    if (t < 32) {
      unsigned long long k = skey[t];
#pragma unroll
      for (int s = 16; s > 0; s >>= 1) {
        unsigned long long o = __shfl_xor(k, s, 32);
        k = o > k ? o : k;
      }
      if (t == 0) swin = (int)(~(unsigned)(k & 0xffffffffull));
    }
    __syncthreads();
    int wv = swin;
    if (t == (wv & 1023)) scent = pc[wv >> 10];
    farthest = wv;
    __syncthreads();
  }

  // Gather group features (faithful to reference: GLOBAL row index < 32768)
  unsigned idx = sgidx[t];
  const float4* src = (const float4*)(features + (long)idx * 64);
  float4* dst = (float4*)(gfeat + ((long)b * NGRP + t) * 64);
#pragma unroll
  for (int i = 0; i < 16; ++i) dst[i] = src[i];
}

// ---------------------------------------------------------------------------
// Kernel 2: generic fp32 WMMA GEMM, one wave per 16x16 C tile.
//   bTrans=1 (NT): C[m,n] = sum_k A[m,k]*B[n,k]   (B row-major [N,K])
//   bTrans=0 (NN): C[m,n] = sum_k A[m,k]*B[k,n]   (B row-major [K,N])
// C = alpha*(A op(B)) + bias[n]
// ---------------------------------------------------------------------------
__global__ __launch_bounds__(32) void wmma_gemm_kernel(
    const float* __restrict__ A, const float* __restrict__ B,
    const float* __restrict__ bias, float* __restrict__ C, int K, int lda,
    int ldb, int ldc, int bTrans, float alpha, long strideA, long strideC) {
  A += (long)blockIdx.z * strideA;
  C += (long)blockIdx.z * strideC;
  const int tm = blockIdx.y * 16;
  const int tn = blockIdx.x * 16;
  const int lane = threadIdx.x;
  const int l = lane & 15;
  const int off = (lane >> 4) << 1;  // K sub-offset 0 or 2 (ISA 16x4 layout)

  v8f acc = {0.f, 0.f, 0.f, 0.f, 0.f, 0.f, 0.f, 0.f};
  const float* arow = A + (long)(tm + l) * lda;
  if (bTrans) {
    const float* brow = B + (long)(tn + l) * ldb;
    for (int k = 0; k < K; k += 4) {
      v2f a = {arow[k + off], arow[k + off + 1]};
      v2f bb = {brow[k + off], brow[k + off + 1]};
      acc = __builtin_amdgcn_wmma_f32_16x16x4_f32(false, a, false, bb,
                                                  (short)0, acc, false, false);
    }
  } else {
    for (int k = 0; k < K; k += 4) {
      v2f a = {arow[k + off], arow[k + off + 1]};
      v2f bb = {B[(long)(k + off) * ldb + tn + l],
                B[(long)(k + off + 1) * ldb + tn + l]};
      acc = __builtin_amdgcn_wmma_f32_16x16x4_f32(false, a, false, bb,
                                                  (short)0, acc, false, false);
    }
  }
  const int cr = tm + ((lane >> 4) << 3);  // rows 0-7 (lanes<16) or 8-15
  const int cc = tn + l;
  const float bv = bias ? bias[cc] : 0.f;
#pragma unroll
  for (int r = 0; r < 8; ++r)
    C[(long)(cr + r) * ldc + cc] = alpha * acc[r] + bv;
}

// ---------------------------------------------------------------------------
// Kernel 3: row softmax (in place), one 256-thread block per row
// ---------------------------------------------------------------------------
__global__ __launch_bounds__(256) void softmax_row_kernel(float* __restrict__ S,
                                                          int n) {
  float* r = S + (long)blockIdx.x * n;
  __shared__ float red[256];
  const int t = threadIdx.x;
  float m = -3.4e38f;
  for (int i = t; i < n; i += 256) m = fmaxf(m, r[i]);
  red[t] = m;
  __syncthreads();
  for (int s = 128; s > 0; s >>= 1) {
    if (t < s) red[t] = fmaxf(red[t], red[t + s]);
    __syncthreads();
  }
  m = red[0];
  __syncthreads();
  float sum = 0.f;
  for (int i = t; i < n; i += 256) {
    float e = expf(r[i] - m);
    r[i] = e;
    sum += e;
  }
  red[t] = sum;
  __syncthreads();
  for (int s = 128; s > 0; s >>= 1) {
    if (t < s) red[t] += red[t + s];
    __syncthreads();
  }
  float inv = 1.0f / red[0];
  for (int i = t; i < n; i += 256) r[i] *= inv;
}

// ---------------------------------------------------------------------------
// Kernel 4: nearest group (exact int argmin, first-index ties) + final add
// ---------------------------------------------------------------------------
__global__ __launch_bounds__(256) void nn_out_kernel(
    const int* __restrict__ coords, const float* __restrict__ features,
    const unsigned* __restrict__ gcoord, const float* __restrict__ ctx,
    float* __restrict__ out) {
  __shared__ unsigned gc[NGRP];
  const int b = blockIdx.x / (NPTS / 256);
  const int p = (blockIdx.x % (NPTS / 256)) * 256 + threadIdx.x;
  for (int i = threadIdx.x; i < NGRP; i += 256) gc[i] = gcoord[b * NGRP + i];
  __syncthreads();

  const long gp = (long)b * NPTS + p;
  const int x = coords[gp * 3 + 0];
  const int y = coords[gp * 3 + 1];
  const int z = coords[gp * 3 + 2];
  int best = 0x7fffffff, bi = 0;
#pragma unroll 8
  for (int g = 0; g < NGRP; ++g) {
    unsigned c = gc[g];
    int dx = x - (int)(c & 1023u);
    int dy = y - (int)((c >> 10) & 1023u);
    int dz = z - (int)(c >> 20);
    int d = dx * dx + dy * dy + dz * dz;
    if (d < best) { best = d; bi = g; }
  }
  const float4* f = (const float4*)(features + gp * 64);
  const float4* cx = (const float4*)(ctx + ((long)b * NGRP + bi) * 64);
  float4* o = (float4*)(out + gp * 64);
#pragma unroll
  for (int i = 0; i < 16; ++i) {
    float4 a = f[i], c2 = cx[i];
    o[i] = make_float4(a.x + c2.x, a.y + c2.y, a.z + c2.z, a.w + c2.w);
  }
}

// ---------------------------------------------------------------------------
extern "C" void kernel_launch(void* const* d_in, const int* in_sizes, int n_in,
                              void* d_out, int out_size, void* d_ws,
                              size_t ws_size, hipStream_t stream) {
  const float* features = (const float*)d_in[0];  // [65536,64]
  const int* coords = (const int*)d_in[1];        // [2,32768,3]
  const float* qkv_w = (const float*)d_in[2];     // [192,64]
  const float* qkv_b = (const float*)d_in[3];     // [192]
  const float* proj_w = (const float*)d_in[4];    // [64,64]
  const float* proj_b = (const float*)d_in[5];    // [64]
  float* out = (float*)d_out;

  char* w = (char*)d_ws;
  int* gidx = (int*)w;           w += 2 * NGRP * sizeof(int);
  unsigned* gcoord = (unsigned*)w; w += 2 * NGRP * sizeof(unsigned);
  float* gfeat = (float*)w;      w += (long)2 * NGRP * 64 * sizeof(float);
  float* qkv = (float*)w;        w += (long)2 * NGRP * 192 * sizeof(float);
  float* ao = (float*)w;         w += (long)2 * NGRP * 64 * sizeof(float);
  float* ctx = (float*)w;        w += (long)2 * NGRP * 64 * sizeof(float);
  float* S = (float*)w;          w += (long)NGRP * NGRP * sizeof(float);

  // 1) FPS + feature gather (one workgroup per batch)
  fps_kernel<<<2, 1024, 0, stream>>>(coords, features, gidx, gcoord, gfeat);

  // 2) QKV projection: [1024,64] @ [192,64]^T + bias -> [1024,192], batched
  wmma_gemm_kernel<<<dim3(192 / 16, NGRP / 16, 2), 32, 0, stream>>>(
      gfeat, qkv_w, qkv_b, qkv, 64, 64, 64, 192, 1, 1.0f,
      (long)NGRP * 64, (long)NGRP * 192);

  // 3) Attention per (batch, head); S buffer reused serially
  for (int b = 0; b < 2; ++b) {
    const float* Qb = qkv + (long)b * NGRP * 192;
    for (int h = 0; h < 4; ++h) {
      // scores = 0.25 * Q_h @ K_h^T   (scale = hd^-0.5 = 0.25, exact)
      wmma_gemm_kernel<<<dim3(NGRP / 16, NGRP / 16), 32, 0, stream>>>(
          Qb + h * 16, Qb + 64 + h * 16, nullptr, S, 16, 192, 192, NGRP, 1,
          0.25f, 0, 0);
      softmax_row_kernel<<<NGRP, 256, 0, stream>>>(S, NGRP);
      // O_h = P @ V_h  -> columns h*16.. of attention output [1024,64]
      wmma_gemm_kernel<<<dim3(1, NGRP / 16), 32, 0, stream>>>(
          S, Qb + 128 + h * 16, nullptr, ao + (long)b * NGRP * 64 + h * 16,
          NGRP, NGRP, 192, 64, 0, 1.0f, 0, 0);
    }
  }

  // 4) Output projection: [1024,64] @ [64,64]^T + bias, batched
  wmma_gemm_kernel<<<dim3(64 / 16, NGRP / 16, 2), 32, 0, stream>>>(
      ao, proj_w, proj_b, ctx, 64, 64, 64, 64, 1, 1.0f, (long)NGRP * 64,
      (long)NGRP * 64);

  // 5) Nearest-group argmin + out = features + ctx[argmin]
  nn_out_kernel<<<(2 * NPTS) / 256, 256, 0, stream>>>(coords, features, gcoord,
                                                      ctx, out);
}